// KCN_57320633533156
// MI455X (gfx1250) — compile-verified
//
#include <hip/hip_runtime.h>
#include <stdint.h>

// Problem constants (match reference)
#define B_Q     4096
#define M_TRAIN 50000
#define KNN     31
#define NNODE   32
#define FDIM    256
#define HDIM    256
#define K0DIM   258     // F + 2
#define XSTRIDE 260     // padded (even -> 8B-aligned ds_load_b64 pairs)
#define GAMMA   0.5f    // 1/(2*length_scale^2)

typedef __attribute__((ext_vector_type(2))) float v2f;
typedef __attribute__((ext_vector_type(8))) float v8f;

#if defined(__AMDGCN__) && __has_builtin(__builtin_amdgcn_global_load_async_to_lds_b32)
#define HAVE_ASYNC_LDS 1
typedef __attribute__((address_space(1))) int* as1_ptr;   // global (HIP prints as __device__ int*)
typedef __attribute__((address_space(3))) int* as3_ptr;   // LDS    (HIP prints as __shared__ int*)
#else
#define HAVE_ASYNC_LDS 0
#endif

__device__ __forceinline__ void wait_async_zero() {
#if defined(__AMDGCN__) && HAVE_ASYNC_LDS
#if __has_builtin(__builtin_amdgcn_s_wait_asynccnt)
    __builtin_amdgcn_s_wait_asynccnt(0);
#else
    asm volatile("s_wait_asynccnt 0" ::: "memory");
#endif
#endif
}

// ---------------------------------------------------------------------------
// Kernel 1: brute-force kNN. One 256-thread workgroup per query.
// Per-thread sorted top-31 candidate list in LDS (u64 keys = d2bits<<32 | idx,
// monotone because d2 >= 0), then 31-round tournament merge via LDS tree min.
// ---------------------------------------------------------------------------
__global__ __launch_bounds__(256) void knn_kernel(
    const float* __restrict__ qxy,   // [B,2]
    const float* __restrict__ txy,   // [M,2]
    int* __restrict__ nb)            // [B,KNN]
{
    __shared__ unsigned long long cand[256 * KNN];  // 62 KB
    __shared__ unsigned long long red[256];

    const int b = blockIdx.x;
    const int t = threadIdx.x;
    const float qx = qxy[2 * b + 0];
    const float qy = qxy[2 * b + 1];

    for (int j = 0; j < KNN; ++j) cand[t * KNN + j] = ~0ULL;

    for (int m = t; m < M_TRAIN; m += 256) {
        float dx = txy[2 * m + 0] - qx;
        float dy = txy[2 * m + 1] - qy;
        float d2 = dx * dx + dy * dy;
        unsigned long long key =
            ((unsigned long long)__float_as_uint(d2) << 32) | (unsigned)m;
        if (key < cand[t * KNN + (KNN - 1)]) {
            int j = KNN - 1;
            while (j > 0 && cand[t * KNN + (j - 1)] > key) {
                cand[t * KNN + j] = cand[t * KNN + (j - 1)];
                --j;
            }
            cand[t * KNN + j] = key;
        }
    }
    __syncthreads();

    int p = 0;
    for (int r = 0; r < KNN; ++r) {
        unsigned long long head = (p < KNN) ? cand[t * KNN + p] : ~0ULL;
        red[t] = head;
        __syncthreads();
        for (int s = 128; s > 0; s >>= 1) {
            if (t < s) {
                unsigned long long o = red[t + s];
                if (o < red[t]) red[t] = o;
            }
            __syncthreads();
        }
        unsigned long long win = red[0];
        if (head == win) ++p;                        // keys are unique (idx in LSBs)
        if (t == 0) nb[b * KNN + r] = (int)(win & 0xffffffffu);
        __syncthreads();
    }
}

// ---------------------------------------------------------------------------
// Fused per-graph GCN: one 256-thread (8 wave32) workgroup per graph.
// f32 WMMA 16x16x4. Tile layout per documented VGPR layouts:
//   A (16x4):  v0 = X[m][k] (lanes0-15) / X[m][k+2] (lanes16-31); v1 = k+1/k+3
//              -> per-lane contiguous pair at k + 2*(lane>>4)  (ds_load_b64)
//   B (4x16):  v0 = W[k][n] / W[k+2][n]; v1 = W[k+1][n] / W[k+3][n]
//   C/D:       vr: lanes0-15 -> (M=r, N=lane), lanes16-31 -> (M=r+8, N=lane-16)
//
// Main K loop is fully unguarded (64 steps = K 0..255). Layer 0's two extra
// rows (256,257 real; 258,259 zero-padded in LDS) are handled by one tail
// step with row indices clamped to Kd-1: the A values for the padded rows
// are exactly 0.0 in LDS, so the clamped (finite) B values contribute 0.
// ---------------------------------------------------------------------------
__device__ __forceinline__ void gcn_layer(
    const float* __restrict__ W, int tail,
    float* __restrict__ Xs, float* __restrict__ Ps, const float* __restrict__ Ah,
    int wave, int lane)
{
    const int lh = lane >> 4;   // 0 or 1 (half-wave)
    const int ll = lane & 15;

    // ---- GEMM1: P[32,256] = X[32,Kd] @ W[Kd,256] ----
    for (int ti = 0; ti < 4; ++ti) {
        const int tile = wave * 4 + ti;       // 0..31
        const int mt = tile & 1;
        const int nt = tile >> 1;
        const int mrow = mt * 16 + ll;
        const int col  = nt * 16 + ll;
        v8f acc = {};
        for (int ks = 0; ks < 64; ++ks) {     // K = 0..255, always in-bounds
            const int k = 4 * ks + 2 * lh;    // even -> 8B aligned pair
            v2f a = *(const v2f*)&Xs[mrow * XSTRIDE + k];
            v2f bb;
            bb.x = W[(size_t)k       * HDIM + col];
            bb.y = W[(size_t)(k + 1) * HDIM + col];
            acc = __builtin_amdgcn_wmma_f32_16x16x4_f32(
                false, a, false, bb, (short)0, acc, false, false);
        }
        if (tail) {                           // layer 0: K rows 256..259
            const int k = 256 + 2 * lh;
            v2f a = *(const v2f*)&Xs[mrow * XSTRIDE + k];   // zeros for lh==1
            const int r0 = (k     < K0DIM) ? k     : (K0DIM - 1);
            const int r1 = (k + 1 < K0DIM) ? k + 1 : (K0DIM - 1);
            v2f bb;
            bb.x = W[(size_t)r0 * HDIM + col];
            bb.y = W[(size_t)r1 * HDIM + col];
            acc = __builtin_amdgcn_wmma_f32_16x16x4_f32(
                false, a, false, bb, (short)0, acc, false, false);
        }
#pragma unroll
        for (int r = 0; r < 8; ++r) {
            const int row = mt * 16 + r + 8 * lh;
            Ps[row * HDIM + col] = acc[r];
        }
    }
    __syncthreads();

    // ---- GEMM2: Xnew[32,256] = relu(Ah[32,32] @ P[32,256]) ----
    for (int ti = 0; ti < 4; ++ti) {
        const int tile = wave * 4 + ti;
        const int mt = tile & 1;
        const int nt = tile >> 1;
        const int mrow = mt * 16 + ll;
        const int col  = nt * 16 + ll;
        v8f acc = {};
        for (int ks = 0; ks < 8; ++ks) {      // K = 32
            const int k = 4 * ks + 2 * lh;
            v2f a = *(const v2f*)&Ah[mrow * NNODE + k];
            v2f bb;
            bb.x = Ps[k       * HDIM + col];
            bb.y = Ps[(k + 1) * HDIM + col];
            acc = __builtin_amdgcn_wmma_f32_16x16x4_f32(
                false, a, false, bb, (short)0, acc, false, false);
        }
#pragma unroll
        for (int r = 0; r < 8; ++r) {
            const int row = mt * 16 + r + 8 * lh;
            Xs[row * XSTRIDE + col] = fmaxf(acc[r], 0.0f);
        }
    }
    __syncthreads();
}

__global__ __launch_bounds__(256) void gcn_kernel(
    const float* __restrict__ qxy,    // [B,2]
    const float* __restrict__ qfeat,  // [B,256]
    const float* __restrict__ txy,    // [M,2]
    const float* __restrict__ tfeat,  // [M,256]
    const float* __restrict__ ty,     // [M,1]
    const float* __restrict__ W0,     // [258,256]
    const float* __restrict__ W1,     // [256,256]
    const float* __restrict__ W2,     // [256,256]
    const float* __restrict__ Wout,   // [256]
    const int*   __restrict__ nb,     // [B,KNN]
    float* __restrict__ out)          // [B]
{
    __shared__ float Xs[NNODE * XSTRIDE];   // 33.3 KB node features / hidden
    __shared__ float Ps[NNODE * HDIM];      // 32 KB   X@W staging
    __shared__ float Ah[NNODE * NNODE];     // 4 KB    normalized adjacency
    __shared__ float cx[NNODE], cy[NNODE];
    __shared__ float dinv[NNODE];
    __shared__ int   nbs[KNN];
    __shared__ float red[256];

    const int b    = blockIdx.x;
    const int t    = threadIdx.x;
    const int lane = t & 31;
    const int wave = t >> 5;

    if (t < KNN) nbs[t] = nb[b * KNN + t];
    // Pull next-layer weights toward L2 while we build the graph.
    __builtin_prefetch(&W1[(size_t)t * 256], 0, 1);
    __builtin_prefetch(&W2[(size_t)t * 256], 0, 1);
    __syncthreads();

    // ---- coords ----
    if (t < NNODE) {
        if (t == 0) { cx[0] = qxy[2 * b]; cy[0] = qxy[2 * b + 1]; }
        else {
            const int m = nbs[t - 1];
            cx[t] = txy[2 * m]; cy[t] = txy[2 * m + 1];
        }
    }

    // ---- gather node features: X[r] = [feat ++ y ++ indicator ++ 0 0] ----
#if HAVE_ASYNC_LDS
    // ASYNCcnt-tracked direct global->LDS copy: each row of 256 floats is one
    // fully coalesced 1KB transfer, no VGPR round-trip; all 32 rows in flight
    // before a single s_wait_asynccnt 0.
    for (int r = 0; r < NNODE; ++r) {
        const float* src = (r == 0) ? &qfeat[(size_t)b * FDIM + t]
                                    : &tfeat[(size_t)nbs[r - 1] * FDIM + t];
        __builtin_amdgcn_global_load_async_to_lds_b32(
            (as1_ptr)(uintptr_t)src,
            (as3_ptr)(uintptr_t)&Xs[r * XSTRIDE + t],
            0, 0);
    }
    wait_async_zero();
#else
    for (int r = 0; r < NNODE; ++r) {
        float v = (r == 0) ? qfeat[(size_t)b * FDIM + t]
                           : tfeat[(size_t)nbs[r - 1] * FDIM + t];
        Xs[r * XSTRIDE + t] = v;
    }
#endif
    if (t < NNODE) {
        const int r = t;
        Xs[r * XSTRIDE + 256] = (r == 0) ? 0.0f : ty[nbs[r - 1]];
        Xs[r * XSTRIDE + 257] = (r == 0) ? 1.0f : 0.0f;
        Xs[r * XSTRIDE + 258] = 0.0f;   // K-padding
        Xs[r * XSTRIDE + 259] = 0.0f;
    }
    __syncthreads();

    // ---- RBF adjacency + GCN symmetric normalization ----
    for (int e = t; e < NNODE * NNODE; e += 256) {
        const int i = e >> 5, j = e & 31;
        const float dx = cx[i] - cx[j], dy = cy[i] - cy[j];
        Ah[e] = expf(-GAMMA * (dx * dx + dy * dy));
    }
    __syncthreads();
    if (t < NNODE) {
        float s = 0.0f;
        for (int j = 0; j < NNODE; ++j) s += Ah[t * NNODE + j];
        dinv[t] = rsqrtf(s);
    }
    __syncthreads();
    for (int e = t; e < NNODE * NNODE; e += 256) {
        const int i = e >> 5, j = e & 31;
        Ah[e] *= dinv[i] * dinv[j];
    }
    __syncthreads();

    // ---- 3x GCN layer (WMMA f32) ----
    gcn_layer(W0, /*tail=*/1, Xs, Ps, Ah, wave, lane);  // Kd = 258 (padded 260)
    gcn_layer(W1, /*tail=*/0, Xs, Ps, Ah, wave, lane);
    gcn_layer(W2, /*tail=*/0, Xs, Ps, Ah, wave, lane);

    // ---- head: out[b] = h[0,:] . Wout ----
    red[t] = Xs[t] * Wout[t];     // row 0 of Xs
    __syncthreads();
    for (int s = 128; s > 0; s >>= 1) {
        if (t < s) red[t] += red[t + s];
        __syncthreads();
    }
    if (t == 0) out[b] = red[0];
}

// ---------------------------------------------------------------------------
extern "C" void kernel_launch(void* const* d_in, const int* in_sizes, int n_in,
                              void* d_out, int out_size, void* d_ws, size_t ws_size,
                              hipStream_t stream) {
    (void)in_sizes; (void)n_in; (void)out_size; (void)ws_size;
    const float* qxy   = (const float*)d_in[0];
    const float* qfeat = (const float*)d_in[1];
    const float* txy   = (const float*)d_in[2];
    const float* tfeat = (const float*)d_in[3];
    const float* ty    = (const float*)d_in[4];
    const float* W0    = (const float*)d_in[5];
    const float* W1    = (const float*)d_in[6];
    const float* W2    = (const float*)d_in[7];
    const float* Wout  = (const float*)d_in[8];

    int* nb = (int*)d_ws;   // 4096*31 ints = 508 KB scratch

    knn_kernel<<<B_Q, 256, 0, stream>>>(qxy, txy, nb);
    gcn_kernel<<<B_Q, 256, 0, stream>>>(qxy, qfeat, txy, tfeat, ty,
                                        W0, W1, W2, Wout, nb, (float*)d_out);
}